// MambaPure_31928786879037
// MI455X (gfx1250) — compile-verified
//
#include <hip/hip_runtime.h>
#include <math.h>

// ---------------------------------------------------------------------------
// Mamba block for MI455X (gfx1250, wave32).
// fp32 end-to-end; big GEMMs use V_WMMA_F32_16X16X4_F32 (native fp32 matrix op).
// Working set (~115MB) is L2-resident (192MB); GEMM AI ~190 FLOP/B.
// Round 2 lesson: ping/pong fragments must be NAMED variables (SSA), not
// arrays indexed by a dynamic buffer id (that lowered to v_cndmask chains).
// ---------------------------------------------------------------------------

typedef float  v2f __attribute__((ext_vector_type(2)));
typedef float  v8f __attribute__((ext_vector_type(8)));

#define B_SZ 2
#define L_SZ 2048
#define DM   768
#define DI   1536
#define DS   16
#define NROW (B_SZ * L_SZ)   // 4096 rows (b*L + l)

// ---------------------------------------------------------------------------
// C[M,N] = A[M,K] @ W[N,K]^T, all row-major. One wave per 32x64 output tile.
// Two M-subtiles share four B fragments: 6 loads feed 8 WMMAs per k-step.
// Software pipeline depth 2 (batches A/B), named registers only.
//
// WMMA f32 16x16x4 register layout (ISA 7.12.2):
//   A: lane = (half<<4)|m ; a[0]=A[m][k0+2*half], a[1]=A[m][k0+2*half+1]
//   B: same K-split across lane halves, N = lane&15
//   C/D: vgpr r -> row (r + 8*half), col = lane&15
// ---------------------------------------------------------------------------
template <int K>
__global__ __launch_bounds__(32) void gemm32x64_f32_wmma(
    const float* __restrict__ A, const float* __restrict__ W,
    float* __restrict__ C, int N64, int lda, int ldw, int ldc)
{
    static_assert(K % 8 == 0, "K must be a multiple of 8");
    const int tile = blockIdx.x;
    const int tm   = tile / N64;        // 32-row tile
    const int tn   = tile % N64;        // 64-col tile
    const int lane = threadIdx.x;       // full wave, EXEC all ones
    const int half = lane >> 4;
    const int lr   = lane & 15;
    const int kh   = 2 * half;          // per-lane K sub-offset

    const float* Ar0 = A + (size_t)(tm * 32 +  0 + lr) * lda;
    const float* Ar1 = A + (size_t)(tm * 32 + 16 + lr) * lda;
    const float* W0  = W + (size_t)(tn * 64 +  0 + lr) * ldw;
    const float* W1  = W + (size_t)(tn * 64 + 16 + lr) * ldw;
    const float* W2  = W + (size_t)(tn * 64 + 32 + lr) * ldw;
    const float* W3  = W + (size_t)(tn * 64 + 48 + lr) * ldw;

    v8f acc00 = {}; v8f acc01 = {}; v8f acc02 = {}; v8f acc03 = {};
    v8f acc10 = {}; v8f acc11 = {}; v8f acc12 = {}; v8f acc13 = {};

    auto ld = [&](int kb, v2f& a0, v2f& a1, v2f& b0, v2f& b1, v2f& b2, v2f& b3) {
        a0 = *(const v2f*)(Ar0 + kb);
        a1 = *(const v2f*)(Ar1 + kb);
        b0 = *(const v2f*)(W0 + kb);
        b1 = *(const v2f*)(W1 + kb);
        b2 = *(const v2f*)(W2 + kb);
        b3 = *(const v2f*)(W3 + kb);
    };
    auto mm8 = [&](v2f a0, v2f a1, v2f b0, v2f b1, v2f b2, v2f b3) {
        acc00 = __builtin_amdgcn_wmma_f32_16x16x4_f32(false, a0, false, b0, (short)0, acc00, false, false);
        acc01 = __builtin_amdgcn_wmma_f32_16x16x4_f32(false, a0, false, b1, (short)0, acc01, false, false);
        acc02 = __builtin_amdgcn_wmma_f32_16x16x4_f32(false, a0, false, b2, (short)0, acc02, false, false);
        acc03 = __builtin_amdgcn_wmma_f32_16x16x4_f32(false, a0, false, b3, (short)0, acc03, false, false);
        acc10 = __builtin_amdgcn_wmma_f32_16x16x4_f32(false, a1, false, b0, (short)0, acc10, false, false);
        acc11 = __builtin_amdgcn_wmma_f32_16x16x4_f32(false, a1, false, b1, (short)0, acc11, false, false);
        acc12 = __builtin_amdgcn_wmma_f32_16x16x4_f32(false, a1, false, b2, (short)0, acc12, false, false);
        acc13 = __builtin_amdgcn_wmma_f32_16x16x4_f32(false, a1, false, b3, (short)0, acc13, false, false);
    };

    // ---- software pipeline, depth 2, named ping/pong registers ----
    v2f a0A, a1A, b0A, b1A, b2A, b3A;
    v2f a0B, a1B, b0B, b1B, b2B, b3B;

    ld(kh, a0A, a1A, b0A, b1A, b2A, b3A);
    for (int k = 0; k < K - 8; k += 8) {
        ld(k + 4 + kh, a0B, a1B, b0B, b1B, b2B, b3B);
        mm8(a0A, a1A, b0A, b1A, b2A, b3A);
        ld(k + 8 + kh, a0A, a1A, b0A, b1A, b2A, b3A);
        mm8(a0B, a1B, b0B, b1B, b2B, b3B);
    }
    ld(K - 4 + kh, a0B, a1B, b0B, b1B, b2B, b3B);
    mm8(a0A, a1A, b0A, b1A, b2A, b3A);
    mm8(a0B, a1B, b0B, b1B, b2B, b3B);

    // Store: vgpr r of each acc -> row (r + 8*half) of its 16-row subtile.
    float* Cb0 = C + (size_t)(tm * 32 +  0 + 8 * half) * ldc + tn * 64 + lr;
    float* Cb1 = C + (size_t)(tm * 32 + 16 + 8 * half) * ldc + tn * 64 + lr;
#pragma unroll
    for (int r = 0; r < 8; ++r) {
        float* Cr0 = Cb0 + (size_t)r * ldc;
        float* Cr1 = Cb1 + (size_t)r * ldc;
        Cr0[0]  = acc00[r];  Cr0[16] = acc01[r];
        Cr0[32] = acc02[r];  Cr0[48] = acc03[r];
        Cr1[0]  = acc10[r];  Cr1[16] = acc11[r];
        Cr1[32] = acc12[r];  Cr1[48] = acc13[r];
    }
}

// ---------------------------------------------------------------------------
// Depthwise causal conv (width 4) + bias + SiLU over x_inner (cols 0..DI-1 of xg)
// ---------------------------------------------------------------------------
__global__ __launch_bounds__(256) void conv_silu_kernel(
    const float* __restrict__ xg, const float* __restrict__ conv_w,
    const float* __restrict__ conv_b, float* __restrict__ x_conv)
{
    int idx = blockIdx.x * 256 + threadIdx.x;
    if (idx >= NROW * DI) return;
    int d   = idx % DI;
    int row = idx / DI;          // b*L + l
    int l   = row % L_SZ;

    float acc = conv_b[d];
    const float* w = conv_w + d * 4;
#pragma unroll
    for (int j = 0; j < 4; ++j) {
        int ls = l - 3 + j;
        if (ls >= 0)
            acc = fmaf(w[j], xg[(size_t)(row - 3 + j) * (2 * DI) + d], acc);
    }
    x_conv[idx] = acc / (1.0f + __expf(-acc));   // silu
}

// ---------------------------------------------------------------------------
// ssm[row, n] = sum_k x_conv[row,k] * W_xproj[n,k]   (N=33: VALU dots, 0.4 GF)
// float4 loads: rows are 6144B / 16B aligned.
// ---------------------------------------------------------------------------
__global__ __launch_bounds__(256) void xproj_kernel(
    const float* __restrict__ x_conv, const float* __restrict__ Wx,
    float* __restrict__ ssm)
{
    int idx = blockIdx.x * 256 + threadIdx.x;
    if (idx >= NROW * 33) return;
    int n   = idx % 33;
    int row = idx / 33;
    const float4* xr = (const float4*)(x_conv + (size_t)row * DI);
    const float4* wr = (const float4*)(Wx + (size_t)n * DI);
    float a0 = 0.f, a1 = 0.f, a2 = 0.f, a3 = 0.f;
    for (int k = 0; k < DI / 4; ++k) {
        float4 xv = xr[k];
        float4 wv = wr[k];
        a0 = fmaf(xv.x, wv.x, a0);
        a1 = fmaf(xv.y, wv.y, a1);
        a2 = fmaf(xv.z, wv.z, a2);
        a3 = fmaf(xv.w, wv.w, a3);
    }
    ssm[idx] = (a0 + a1) + (a2 + a3);
}

// ---------------------------------------------------------------------------
// Fused selective scan: one lane per (b, d) channel, h[16] in registers.
// dt softplus + discretization + C contraction + D skip + SiLU(gate) fused.
// ---------------------------------------------------------------------------
__global__ __launch_bounds__(256) void scan_kernel(
    const float* __restrict__ ssm, const float* __restrict__ x_conv,
    const float* __restrict__ xg, const float* __restrict__ w_dt,
    const float* __restrict__ b_dt, const float* __restrict__ A_log,
    const float* __restrict__ D_param, float* __restrict__ y)
{
    int g = blockIdx.x * 256 + threadIdx.x;
    if (g >= B_SZ * DI) return;
    int b = g / DI, d = g % DI;

    const float wdt = w_dt[d];
    const float bdt = b_dt[d];
    const float Dp  = D_param[d];
    float Aa[DS];
#pragma unroll
    for (int n = 0; n < DS; ++n) Aa[n] = -__expf(A_log[d * DS + n]);
    float h[DS];
#pragma unroll
    for (int n = 0; n < DS; ++n) h[n] = 0.f;

    for (int l = 0; l < L_SZ; ++l) {
        const int row = b * L_SZ + l;
        const float* s = ssm + (size_t)row * 33;       // [B(16) | C(16) | dt_raw]
        const float x_t = x_conv[(size_t)row * DI + d];

        float z  = fmaf(s[32], wdt, bdt);
        float dt = (z > 20.0f) ? z : log1pf(__expf(z)); // softplus

        float yv = 0.f;
#pragma unroll
        for (int n = 0; n < DS; ++n) {
            float Abar = __expf(dt * Aa[n]);
            h[n] = fmaf(Abar, h[n], (dt * s[n]) * x_t);
            yv   = fmaf(h[n], s[DS + n], yv);
        }
        yv = fmaf(x_t, Dp, yv);

        float gv = xg[(size_t)row * (2 * DI) + DI + d]; // gate
        float sg = gv / (1.0f + __expf(-gv));           // silu(gate)
        y[(size_t)row * DI + d] = yv * sg;
    }
}

// ---------------------------------------------------------------------------
extern "C" void kernel_launch(void* const* d_in, const int* in_sizes, int n_in,
                              void* d_out, int out_size, void* d_ws, size_t ws_size,
                              hipStream_t stream)
{
    const float* x       = (const float*)d_in[0];
    const float* W_in    = (const float*)d_in[1];
    const float* conv_w  = (const float*)d_in[2];
    const float* conv_b  = (const float*)d_in[3];
    const float* W_xproj = (const float*)d_in[4];
    const float* w_dt    = (const float*)d_in[5];
    const float* b_dt    = (const float*)d_in[6];
    const float* A_log   = (const float*)d_in[7];
    const float* D_param = (const float*)d_in[8];
    const float* W_out   = (const float*)d_in[9];
    float* out = (float*)d_out;

    // workspace carve-up (floats): xg | x_conv | ssm | y   (~101 MB total)
    float* ws     = (float*)d_ws;
    float* xg     = ws;                              // NROW x 3072
    float* x_conv = xg + (size_t)NROW * (2 * DI);    // NROW x 1536
    float* ssm    = x_conv + (size_t)NROW * DI;      // NROW x 33
    float* yb     = ssm + (size_t)NROW * 33;         // NROW x 1536

    // 1) xg = x @ W_in^T   (M=4096, N=3072, K=768)
    gemm32x64_f32_wmma<DM><<<(NROW / 32) * ((2 * DI) / 64), 32, 0, stream>>>(
        x, W_in, xg, (2 * DI) / 64, DM, DM, 2 * DI);

    // 2) depthwise conv + bias + silu
    conv_silu_kernel<<<(NROW * DI + 255) / 256, 256, 0, stream>>>(
        xg, conv_w, conv_b, x_conv);

    // 3) ssm = x_conv @ W_xproj^T  (N=33)
    xproj_kernel<<<(NROW * 33 + 255) / 256, 256, 0, stream>>>(
        x_conv, W_xproj, ssm);

    // 4) fused selective scan + gating
    scan_kernel<<<(B_SZ * DI + 255) / 256, 256, 0, stream>>>(
        ssm, x_conv, xg, w_dt, b_dt, A_log, D_param, yb);

    // 5) out = y @ W_out^T  (M=4096, N=768, K=1536)
    gemm32x64_f32_wmma<DI><<<(NROW / 32) * (DM / 64), 32, 0, stream>>>(
        yb, W_out, out, DM / 64, DI, DI, DM);
}